// SeqExperts_81990925680846
// MI455X (gfx1250) — compile-verified
//
#include <hip/hip_runtime.h>
#include <hip/hip_bf16.h>

// D = A(16x4 f32) x B(4x16 f32) + C(16x16 f32), wave32 WMMA.
typedef __attribute__((ext_vector_type(2))) float v2f;
typedef __attribute__((ext_vector_type(8))) float v8f;

#define N_TOK   131072
#define NE      8
#define DD      128
#define T_CHUNK 64              // tokens staged per block iteration
#define PITCH   132             // LDS row pitch (floats): 132 mod 64 = 4 -> conflict-free b64 reads
#define BPE     512             // blocks per expert
#define CHUNKS_PER_EXPERT (N_TOK / T_CHUNK)   // 2048
#define ITERS   (CHUNKS_PER_EXPERT / BPE)     // 4

__global__ __launch_bounds__(256)
void seq_experts_wmma_f32(const float* __restrict__ in,
                          const float* __restrict__ W,
                          const float* __restrict__ Bias,
                          float* __restrict__ out) {
  __shared__ float sA[T_CHUNK * PITCH];       // 64 x 132 fp32 = 33 KB

  const int tid  = threadIdx.x;
  const int wave = tid >> 5;                  // 0..7  -> 16-wide feature stripe
  const int lane = tid & 31;
  const int half = lane >> 4;                 // 0/1 (lane group)
  const int m    = lane & 15;                 // A row / B column within tile

  const int e  = blockIdx.x / BPE;            // expert
  const int c0 = blockIdx.x % BPE;            // first chunk index

  const int fcol = wave * 16 + m;             // this lane's output feature column

  // ---- B fragments: W[e][:, fcol] resident in 64 VGPRs for the whole block ----
  // Assumed WMMA f32 16x16x4 B layout (mirror of documented A layout):
  //   VGPR0: lanes0-15 = B[k+0][n], lanes16-31 = B[k+2][n]
  //   VGPR1: lanes0-15 = B[k+1][n], lanes16-31 = B[k+3][n]
  const float* We = W + (size_t)e * DD * DD;
  v2f bfrag[32];
#pragma unroll
  for (int ks = 0; ks < 32; ++ks) {
    const int k = 4 * ks + 2 * half;
    bfrag[ks].x = We[(size_t)k * DD + fcol];
    bfrag[ks].y = We[(size_t)(k + 1) * DD + fcol];
  }
  const float bias = Bias[e * DD + fcol];

  for (int j = 0; j < ITERS; ++j) {
    const long chunk = c0 + j * BPE;
    const long tok0  = chunk * T_CHUNK;

    // ---- stage A chunk (64 tokens x 128 d), coalesced global_load_b128 ----
#pragma unroll
    for (int i = 0; i < (T_CHUNK * DD / 4) / 256; ++i) {   // 8 iterations
      const int q   = i * 256 + tid;
      const int row = q >> 5;                 // 0..63
      const int qd  = (q & 31) * 4;           // d offset (float4 granule)
      const float4 v =
          *(const float4*)&in[((size_t)(tok0 + row) * NE + e) * DD + qd];
      *(float4*)&sA[row * PITCH + qd] = v;
    }
    // speculative prefetch of next chunk's rows (global_prefetch_b8)
    if (j + 1 < ITERS) {
      __builtin_prefetch(
          &in[((size_t)(tok0 + BPE * (long)T_CHUNK + (tid >> 2)) * NE + e) * DD],
          0, 1);
    }
    __syncthreads();

    // ---- 4 token tiles of 16: 32 x (ds_load_b64 + v_wmma_f32_16x16x4_f32) ----
    for (int t = 0; t < T_CHUNK / 16; ++t) {
      v8f c = {};
      const float* aRow = &sA[(t * 16 + m) * PITCH + 2 * half];
#pragma unroll
      for (int ks = 0; ks < 32; ++ks) {
        const v2f a = *(const v2f*)&aRow[4 * ks];  // A[m][4ks+2h], A[m][4ks+2h+1]
        c = __builtin_amdgcn_wmma_f32_16x16x4_f32(
                /*neg_a=*/false, a, /*neg_b=*/false, bfrag[ks],
                /*c_mod=*/(short)0, c, /*reuse_a=*/false, /*reuse_b=*/false);
      }
      // C/D layout: VGPR r -> token row (r + 8*half), N = lane&15 = fcol stripe
      const long tokBase = tok0 + t * 16 + 8 * half;
#pragma unroll
      for (int r = 0; r < 8; ++r) {
        out[((size_t)(tokBase + r) * NE + e) * DD + fcol] = c[r] + bias;
      }
    }
    __syncthreads();
  }
}

extern "C" void kernel_launch(void* const* d_in, const int* in_sizes, int n_in,
                              void* d_out, int out_size, void* d_ws, size_t ws_size,
                              hipStream_t stream) {
  (void)in_sizes; (void)n_in; (void)d_ws; (void)ws_size; (void)out_size;
  const float* in   = (const float*)d_in[0];   // [N, E, D] fp32
  const float* W    = (const float*)d_in[1];   // [E, D, D] fp32
  const float* bias = (const float*)d_in[2];   // [E, D]    fp32
  float* out        = (float*)d_out;           // [N, E, D] fp32

  dim3 grid(NE * BPE);   // 4096 blocks, each: 1 expert, 4 x 64-token chunks
  dim3 block(256);       // 8 waves (wave32)
  seq_experts_wmma_f32<<<grid, block, 0, stream>>>(in, W, bias, out);
}